// StripedConv_74345883894469
// MI455X (gfx1250) — compile-verified
//
#include <hip/hip_runtime.h>

// ---------------------------------------------------------------------------
// MI455X (gfx1250, wave32). fp32 matrix work -> V_WMMA_F32_16X16X4_F32.
// Fragment layouts per CDNA5 ISA 7.12.2:
//   A (16x4 f32):  lane L: M = L&15 ; comp0 = k0+((L>>4)*2), comp1 = comp0+1
//   B (4x16 f32):  lane L: N = L&15 ; comp0 = row k0+((L>>4)*2), comp1 = +1
//   C/D (16x16):   VGPR i: M = i + ((L>>4)*8), N = L&15
// Weights are repacked once into A-fragment order so each A load is one b64.
// Input tiles are staged with GLOBAL_LOAD_ASYNC_TO_LDS_B128 (ASYNCcnt).
// ---------------------------------------------------------------------------

typedef float v2f __attribute__((ext_vector_type(2)));
typedef float v8f __attribute__((ext_vector_type(8)));

#define HWSZ 65536              // 256*256
#define NHWF 262144.0f          // N*H*W for BN statistics
#define PLANE_F 16777216        // 4*64*256*256

__device__ __forceinline__ v8f wmma_f32(v2f a, v2f b, v8f c) {
  return __builtin_amdgcn_wmma_f32_16x16x4_f32(
      false, a, false, b, (short)0, c, false, false);
}

// A-fragment from packed weights: pack[(k>>1)*64 + oc] = {w(k), w(k+1)}
__device__ __forceinline__ v2f loadA_pack(const float2* __restrict__ wp, int kx, int row) {
  float2 t = wp[((kx >> 1) << 6) + row];
  v2f a; a.x = t.x; a.y = t.y; return a;
}
// B-fragment, vertical conv: k = dr*64 + ic ; LDS in[ic][rbase+dr][col]
__device__ __forceinline__ v2f loadB_vert(const float* __restrict__ s, int kx, int rbase, int colN) {
  int ic0 = kx & 63,       dr0 = kx >> 6;
  int ic1 = (kx + 1) & 63, dr1 = (kx + 1) >> 6;
  v2f b;
  b.x = s[ic0 * 240 + (rbase + dr0) * 48 + colN];
  b.y = s[ic1 * 240 + (rbase + dr1) * 48 + colN];
  return b;
}
// B-fragment, horizontal conv: k = dc*64 + mc ; tile[mc][48 cols]
__device__ __forceinline__ v2f loadB_hor(const float* __restrict__ t, int kx, int cbase, int N) {
  int mc0 = kx & 63,       dc0 = kx >> 6;
  int mc1 = (kx + 1) & 63, dc1 = (kx + 1) >> 6;
  v2f b;
  b.x = t[mc0 * 48 + cbase + dc0 + N];
  b.y = t[mc1 * 48 + cbase + dc1 + N];
  return b;
}
// B-fragment, pointwise: dw_s[k][16 px]
__device__ __forceinline__ v2f loadB_pw(const float* __restrict__ d, int kx, int N) {
  v2f b; b.x = d[kx * 16 + N]; b.y = d[(kx + 1) * 16 + N]; return b;
}

// Software-pipelined K-loop: prefetch distance 2 fragment pairs, 2 WMMA / body.
#define GEMM_PIPE(K, LA, LB, ACC)                                   \
  {                                                                 \
    v2f A0 = LA(khalf),     B0 = LB(khalf);                         \
    v2f A1 = LA(4 + khalf), B1 = LB(4 + khalf);                     \
    for (int k0 = 0; k0 < (K); k0 += 8) {                           \
      int kn0 = (k0 + 8  < (K)) ? k0 + 8  + khalf : khalf;          \
      int kn1 = (k0 + 12 < (K)) ? k0 + 12 + khalf : khalf;          \
      v2f A2 = LA(kn0), B2 = LB(kn0);                               \
      v2f A3 = LA(kn1), B3 = LB(kn1);                               \
      ACC = wmma_f32(A0, B0, ACC);                                  \
      ACC = wmma_f32(A1, B1, ACC);                                  \
      A0 = A2; B0 = B2; A1 = A3; B1 = B3;                           \
    }                                                               \
  }

// Async global -> LDS (128b), tracked by ASYNCcnt (CDNA5 §15.18.3 op 98)
__device__ __forceinline__ void async_load_lds_b128(const float* base, int byte_off, void* lds_dst) {
  unsigned l = (unsigned)(size_t)lds_dst;   // flat addr low 32 bits == LDS offset
  asm volatile("global_load_async_to_lds_b128 %0, %1, %2"
               :: "v"(l), "v"(byte_off), "s"(base) : "memory");
}
__device__ __forceinline__ void wait_asynccnt0() {
  asm volatile("s_wait_asynccnt 0x0" ::: "memory");
}

struct BranchW {
  const float2 *p1, *p31, *p13, *p51, *p15;   // packed weights
  const float  *b1, *b31, *b13, *b51, *b15;   // biases
};

// ---------------------------------------------------------------------------
// K0a: zero the atomic-accumulated BN stats region
// ---------------------------------------------------------------------------
__global__ void zero_stats_kernel(float* stats) {
  for (int i = blockIdx.x * 256 + threadIdx.x; i < 1536; i += gridDim.x * 256)
    stats[i] = 0.0f;
}

// ---------------------------------------------------------------------------
// K0b: repack conv weights into A-fragment order.
// src[(oc*IC + ic)*taps + dr], k = dr*IC + ic  ->  dst[(k>>1)*64 + oc]
// ---------------------------------------------------------------------------
__global__ void pack_weights_kernel(const float* __restrict__ src,
                                    float2* __restrict__ dst, int IC, int taps) {
  int K = IC * taps;
  int tot = 64 * (K >> 1);
  for (int t = blockIdx.x * 256 + threadIdx.x; t < tot; t += gridDim.x * 256) {
    int kp = t >> 6;
    int oc = t & 63;
    int k0 = kp * 2, k1 = k0 + 1;
    float2 v;
    v.x = src[(oc * IC + (k0 % IC)) * taps + (k0 / IC)];
    v.y = src[(oc * IC + (k1 % IC)) * taps + (k1 / IC)];
    dst[t] = v;
  }
}

// ---------------------------------------------------------------------------
// K1: 3x3 dilate (max) / erode (min); border taps never win (skip OOB)
// ---------------------------------------------------------------------------
__global__ void dilate_erode_kernel(const float* __restrict__ x,
                                    float* __restrict__ dil,
                                    float* __restrict__ ero) {
  int i = blockIdx.x * 256 + threadIdx.x;
  int w = i & 255;
  int h = (i >> 8) & 255;
  int base = (i >> 16) << 16;
  float mx = -__builtin_huge_valf();
  float mn =  __builtin_huge_valf();
#pragma unroll
  for (int dr = -1; dr <= 1; ++dr) {
    int r = h + dr;
    if ((unsigned)r >= 256u) continue;
#pragma unroll
    for (int dc = -1; dc <= 1; ++dc) {
      int c = w + dc;
      if ((unsigned)c >= 256u) continue;
      float v = x[base + r * 256 + c];
      mx = fmaxf(mx, v);
      mn = fminf(mn, v);
    }
  }
  dil[i] = mx;
  ero[i] = mn;
}

// ---------------------------------------------------------------------------
// K2: fused branch. Block = (n, h, 32-px tile). 384 threads = 12 waves.
// Stage 1: x1/x31/x51 over 48-px haloed strip (WMMA K=64/192/320)
// Stage 2: x3113/x5115 (WMMA K=192/320) + x1 store + difference partial
// ---------------------------------------------------------------------------
__global__ void __launch_bounds__(384)
branch_kernel(const float* __restrict__ in, float* __restrict__ cat,
              float* __restrict__ diff, BranchW p, int isE) {
  int bx = blockIdx.x;
  int wt = bx & 7;
  int h  = (bx >> 3) & 255;
  int n  = bx >> 11;
  int w0 = wt * 32;

  int tid  = threadIdx.x;
  int lane = tid & 31;
  int wid  = tid >> 5;

  __shared__ float smem[15360];   // in[64ch][5row][48col]; later x31/x51/x1 tiles

  // ---- async stage input tile: rows h-2..h+2, cols w0-8..w0+39, zero pad ----
  for (int e4 = tid; e4 < 64 * 5 * 12; e4 += 384) {
    int ic  = e4 / 60;
    int rem = e4 - ic * 60;
    int r   = rem / 12;
    int c4  = (rem - r * 12) * 4;
    int gr  = h - 2 + r;
    int gc  = w0 - 8 + c4;
    float* ldst = &smem[ic * 240 + r * 48 + c4];
    bool rowok = (unsigned)gr < 256u;
    if (rowok && gc >= 0 && gc + 3 < 256) {
      int goff = ((((n * 64 + ic) * 256 + gr) * 256) + gc) * 4;
      async_load_lds_b128(in, goff, ldst);
    } else {
#pragma unroll
      for (int j = 0; j < 4; ++j) {
        int gcj = gc + j;
        ldst[j] = (rowok && (unsigned)gcj < 256u)
                    ? in[((n * 64 + ic) * 256 + gr) * 256 + gcj] : 0.0f;
      }
    }
  }
  wait_asynccnt0();
  __syncthreads();

  // ---- stage 1: vertical convs ----
  int ot    = wid & 3;            // oc tile 0..3
  int pt    = wid >> 2;           // col subtile 0..2
  int ocb   = ot * 16;
  int col0  = pt * 16;
  int M     = lane & 15;
  int N     = M;
  int khalf = (lane >> 4) << 1;   // 0 or 2
  int rsel  = (lane >> 4) << 3;   // C/D row select
  int ocbM  = ocb + M;
  int colN  = col0 + N;

  v8f a1 = {}, a31 = {}, a51 = {};

#define LA1(kx)  loadA_pack(p.p1,  (kx), ocbM)
#define LB1(kx)  loadB_vert(smem, (kx), 2, colN)
  GEMM_PIPE(64, LA1, LB1, a1)
#undef LA1
#undef LB1

#define LA31(kx) loadA_pack(p.p31, (kx), ocbM)
#define LB31(kx) loadB_vert(smem, (kx), 1, colN)
  GEMM_PIPE(192, LA31, LB31, a31)
#undef LA31
#undef LB31

#define LA51(kx) loadA_pack(p.p51, (kx), ocbM)
#define LB51(kx) loadB_vert(smem, (kx), 0, colN)
  GEMM_PIPE(320, LA51, LB51, a51)
#undef LA51
#undef LB51

  __syncthreads();

  // ---- overwrite LDS with x31/x51/x1 tiles (bias added; OOB cols -> 0) ----
  float* x31s = smem;
  float* x51s = smem + 3072;
  float* x1s  = smem + 6144;
  {
    int gc  = w0 - 8 + colN;
    bool ib = (unsigned)gc < 256u;
#pragma unroll
    for (int i = 0; i < 8; ++i) {
      int oc = ocb + i + rsel;
      x1s [oc * 48 + colN] = ib ? (a1[i]  + p.b1[oc])  : 0.0f;
      x31s[oc * 48 + colN] = ib ? (a31[i] + p.b31[oc]) : 0.0f;
      x51s[oc * 48 + colN] = ib ? (a51[i] + p.b51[oc]) : 0.0f;
    }
  }
  __syncthreads();

  // ---- stage 2 ----
  if (wid < 8) {
    int ot2  = wid & 3;
    int ct   = wid >> 2;          // 0..1
    int ocb2 = ot2 * 16;
    int ocb2M = ocb2 + M;
    int cb   = 8 + ct * 16;       // tile-local col base of outputs
    v8f acc13 = {}, acc15 = {};

#define LA13(kx) loadA_pack(p.p13, (kx), ocb2M)
#define LB13(kx) loadB_hor(x31s, (kx), cb - 1, N)
    GEMM_PIPE(192, LA13, LB13, acc13)
#undef LA13
#undef LB13

#define LA15(kx) loadA_pack(p.p15, (kx), ocb2M)
#define LB15(kx) loadB_hor(x51s, (kx), cb - 2, N)
    GEMM_PIPE(320, LA15, LB15, acc15)
#undef LA15
#undef LB15

    int basec = (n * 192) * HWSZ + h * 256 + w0 + ct * 16 + N;
#pragma unroll
    for (int i = 0; i < 8; ++i) {
      int oc = ocb2 + i + rsel;
      cat[basec + (64  + oc) * HWSZ] = acc13[i] + p.b13[oc];
      cat[basec + (128 + oc) * HWSZ] = acc15[i] + p.b15[oc];
    }
  } else {
    // waves 8..11: store x1 (cat ch [0,64)) and the difference partial
    int ot2 = wid - 8;
    for (int idx = lane; idx < 512; idx += 32) {
      int ocl = idx >> 5;
      int cl  = idx & 31;
      int oc  = ot2 * 16 + ocl;
      float v1  = x1s [oc * 48 + 8 + cl];
      float v31 = x31s[oc * 48 + 8 + cl];
      float v51 = x51s[oc * 48 + 8 + cl];
      cat[((n * 192 + oc) * 256 + h) * 256 + w0 + cl] = v1;
      float dsum = 2.0f * v1 + v31 + v51;
      int didx = ((n * 64 + oc) * 256 + h) * 256 + w0 + cl;
      diff[didx] = isE ? (diff[didx] - dsum) : dsum;
    }
  }
}

// ---------------------------------------------------------------------------
// K3: per-channel sum / sumsq of Dcat & Ecat (atomic block partials)
// stats: [0:192) sumD [192:384) sqD [384:576) sumE [576:768) sqE
//        [768:960) scD [960:1152) shD [1152:1344) scE [1344:1536) shE
// ---------------------------------------------------------------------------
__global__ void __launch_bounds__(256)
stats_kernel(const float* __restrict__ Dcat, const float* __restrict__ Ecat,
             float* __restrict__ stats) {
  int b     = blockIdx.x;        // 6144 blocks
  int chunk = b & 3;
  int plane = b >> 2;            // 0..1535
  int tens  = plane >= 768;
  int rem   = plane - tens * 768;
  int c     = rem % 192;
  const float* src = tens ? Ecat : Dcat;
  int base = rem * HWSZ + chunk * 16384;

  float s = 0.0f, q = 0.0f;
  for (int j = threadIdx.x; j < 16384; j += 256) {
    float v = src[base + j];
    s += v;
    q += v * v;
  }
  __shared__ float rs[256], rq[256];
  int tid = threadIdx.x;
  rs[tid] = s; rq[tid] = q;
  __syncthreads();
  for (int off = 128; off > 0; off >>= 1) {
    if (tid < off) { rs[tid] += rs[tid + off]; rq[tid] += rq[tid + off]; }
    __syncthreads();
  }
  if (tid == 0) {
    atomicAdd(&stats[tens * 384 + c],       rs[0]);
    atomicAdd(&stats[tens * 384 + 192 + c], rq[0]);
  }
}

// ---------------------------------------------------------------------------
// K3b: fold BN stats to per-channel scale/shift
// ---------------------------------------------------------------------------
__global__ void bn_finalize_kernel(float* stats, const float* gD, const float* bD,
                                   const float* gE, const float* bE) {
  int i = threadIdx.x;
  if (i >= 384) return;
  int tens = i >= 192;
  int c    = i - tens * 192;
  float s    = stats[tens * 384 + c];
  float q    = stats[tens * 384 + 192 + c];
  float mean = s / NHWF;
  float var  = q / NHWF - mean * mean;
  float g    = tens ? gE[c] : gD[c];
  float be   = tens ? bE[c] : bD[c];
  float sc   = g * rsqrtf(var + 1e-5f);
  stats[768 + tens * 384 + c]       = sc;
  stats[768 + tens * 384 + 192 + c] = be - mean * sc;
}

// ---------------------------------------------------------------------------
// K4: BN+ReLU(d) - BN+ReLU(e) -> depthwise 3x3 -> pointwise 192->64 (WMMA)
// Block = (n, h, 16-px tile). 384 threads = 12 waves.
// ---------------------------------------------------------------------------
__global__ void __launch_bounds__(384)
edge_kernel(const float* __restrict__ Dcat, const float* __restrict__ Ecat,
            const float* __restrict__ stats,
            const float* __restrict__ dww, const float* __restrict__ dwb,
            const float2* __restrict__ pwp, const float* __restrict__ pwb,
            float* __restrict__ edge) {
  int bx = blockIdx.x;
  int wt = bx & 15;
  int h  = (bx >> 4) & 255;
  int n  = bx >> 12;
  int w0 = wt * 16;

  int tid  = threadIdx.x;
  int lane = tid & 31;
  int wid  = tid >> 5;

  __shared__ float t_s[192 * 3 * 20];   // (d-e): rows h-1..h+1, cols w0-1..w0+16
  __shared__ float dw_s[192 * 16];

  const float* scD = stats + 768;
  const float* shD = stats + 960;
  const float* scE = stats + 1152;
  const float* shE = stats + 1344;

  for (int e = tid; e < 192 * 3 * 18; e += 384) {
    int c   = e / 54;
    int rem = e - c * 54;
    int r   = rem / 18;
    int col = rem - r * 18;
    int gr  = h - 1 + r;
    int gc  = w0 - 1 + col;
    float t = 0.0f;
    if ((unsigned)gr < 256u && (unsigned)gc < 256u) {
      int gi   = ((n * 192 + c) * 256 + gr) * 256 + gc;
      float dv = fmaxf(Dcat[gi] * scD[c] + shD[c], 0.0f);
      float ev = fmaxf(Ecat[gi] * scE[c] + shE[c], 0.0f);
      t = dv - ev;
    }
    t_s[(c * 3 + r) * 20 + col] = t;
  }
  __syncthreads();

  for (int e = tid; e < 192 * 16; e += 384) {
    int c = e >> 4;
    int pp = e & 15;
    float acc = dwb[c];
#pragma unroll
    for (int dr = 0; dr < 3; ++dr)
#pragma unroll
      for (int dc = 0; dc < 3; ++dc)
        acc += dww[c * 9 + dr * 3 + dc] * t_s[(c * 3 + dr) * 20 + pp + dc];
    dw_s[e] = acc;
  }
  __syncthreads();

  if (wid < 4) {
    int ocb   = wid * 16;
    int M     = lane & 15;
    int N     = M;
    int khalf = (lane >> 4) << 1;
    int rsel  = (lane >> 4) << 3;
    int ocbM  = ocb + M;
    v8f acc = {};
#define LAp(kx) loadA_pack(pwp, (kx), ocbM)
#define LBp(kx) loadB_pw(dw_s, (kx), N)
    GEMM_PIPE(192, LAp, LBp, acc)
#undef LAp
#undef LBp
#pragma unroll
    for (int i = 0; i < 8; ++i) {
      int oc  = ocb + i + rsel;
      float v = fmaxf(acc[i] + pwb[oc], 0.0f);
      edge[((n * 64 + oc) * 256 + h) * 256 + w0 + N] = v;
    }
  }
}

// ---------------------------------------------------------------------------
// launch
// ---------------------------------------------------------------------------
extern "C" void kernel_launch(void* const* d_in, const int* in_sizes, int n_in,
                              void* d_out, int out_size, void* d_ws, size_t ws_size,
                              hipStream_t stream) {
  const float* x = (const float*)d_in[0];
  const float* Dw1  = (const float*)d_in[1];   const float* Db1  = (const float*)d_in[2];
  const float* Dw31 = (const float*)d_in[3];   const float* Db31 = (const float*)d_in[4];
  const float* Dw13 = (const float*)d_in[5];   const float* Db13 = (const float*)d_in[6];
  const float* Dw51 = (const float*)d_in[7];   const float* Db51 = (const float*)d_in[8];
  const float* Dw15 = (const float*)d_in[9];   const float* Db15 = (const float*)d_in[10];
  const float* Dg   = (const float*)d_in[11];  const float* Db   = (const float*)d_in[12];
  const float* Ew1  = (const float*)d_in[13];  const float* Eb1  = (const float*)d_in[14];
  const float* Ew31 = (const float*)d_in[15];  const float* Eb31 = (const float*)d_in[16];
  const float* Ew13 = (const float*)d_in[17];  const float* Eb13 = (const float*)d_in[18];
  const float* Ew51 = (const float*)d_in[19];  const float* Eb51 = (const float*)d_in[20];
  const float* Ew15 = (const float*)d_in[21];  const float* Eb15 = (const float*)d_in[22];
  const float* Eg   = (const float*)d_in[23];  const float* Eb   = (const float*)d_in[24];
  const float* dww  = (const float*)d_in[25];  const float* dwb  = (const float*)d_in[26];
  const float* pww  = (const float*)d_in[27];  const float* pwb  = (const float*)d_in[28];

  float* ws    = (float*)d_ws;
  float* dil   = ws;
  float* ero   = dil  + PLANE_F;
  float* Dcat  = ero  + PLANE_F;
  float* Ecat  = Dcat + 3 * PLANE_F;
  float* stats = Ecat + 3 * PLANE_F;          // 1536 floats
  float* pk    = stats + 1536;                // packed weights

  float2* pD1  = (float2*)(pk);               //  4096 floats
  float2* pD31 = (float2*)(pk + 4096);        // 12288
  float2* pD13 = (float2*)(pk + 16384);       // 12288
  float2* pD51 = (float2*)(pk + 28672);       // 20480
  float2* pD15 = (float2*)(pk + 49152);       // 20480
  float2* pE1  = (float2*)(pk + 69632);
  float2* pE31 = (float2*)(pk + 73728);
  float2* pE13 = (float2*)(pk + 86016);
  float2* pE51 = (float2*)(pk + 98304);
  float2* pE15 = (float2*)(pk + 118784);
  float2* pPW  = (float2*)(pk + 139264);      // 12288

  float* edge = (float*)d_out;
  float* diff = edge + PLANE_F;

  zero_stats_kernel<<<6, 256, 0, stream>>>(stats);
  pack_weights_kernel<<<8,  256, 0, stream>>>(Dw1,  pD1,  64, 1);
  pack_weights_kernel<<<24, 256, 0, stream>>>(Dw31, pD31, 64, 3);
  pack_weights_kernel<<<24, 256, 0, stream>>>(Dw13, pD13, 64, 3);
  pack_weights_kernel<<<40, 256, 0, stream>>>(Dw51, pD51, 64, 5);
  pack_weights_kernel<<<40, 256, 0, stream>>>(Dw15, pD15, 64, 5);
  pack_weights_kernel<<<8,  256, 0, stream>>>(Ew1,  pE1,  64, 1);
  pack_weights_kernel<<<24, 256, 0, stream>>>(Ew31, pE31, 64, 3);
  pack_weights_kernel<<<24, 256, 0, stream>>>(Ew13, pE13, 64, 3);
  pack_weights_kernel<<<40, 256, 0, stream>>>(Ew51, pE51, 64, 5);
  pack_weights_kernel<<<40, 256, 0, stream>>>(Ew15, pE15, 64, 5);
  pack_weights_kernel<<<24, 256, 0, stream>>>(pww,  pPW, 192, 1);

  dilate_erode_kernel<<<65536, 256, 0, stream>>>(x, dil, ero);

  BranchW D { pD1, pD31, pD13, pD51, pD15, Db1, Db31, Db13, Db51, Db15 };
  BranchW E { pE1, pE31, pE13, pE51, pE15, Eb1, Eb31, Eb13, Eb51, Eb15 };
  branch_kernel<<<8192, 384, 0, stream>>>(dil, Dcat, diff, D, 0);
  branch_kernel<<<8192, 384, 0, stream>>>(ero, Ecat, diff, E, 1);

  stats_kernel<<<6144, 256, 0, stream>>>(Dcat, Ecat, stats);
  bn_finalize_kernel<<<1, 384, 0, stream>>>(stats, Dg, Db, Eg, Eb);
  edge_kernel<<<16384, 384, 0, stream>>>(Dcat, Ecat, stats, dww, dwb, pPW, pwb, edge);
}